// SyntaxAwarePositionalEmbedding_75076028334452
// MI455X (gfx1250) — compile-verified
//
#include <hip/hip_runtime.h>

#define B_DIM 64
#define S_DIM 2048
#define H_DIM 512
#define K_DIM 512

typedef __attribute__((ext_vector_type(2))) float v2f;
typedef __attribute__((ext_vector_type(8))) float v8f;

// ---------------------------------------------------------------------------
// Kernel 1: D[M x 512] = A[M x 512] * Bm[512 x 512], fp32 WMMA 16x16x4.
// One wave32 per 16x16 output tile. A rows are clamped so tiny tables
// (M=16, M=8) reuse the same kernel; only rows < M are stored.
//
// VGPR layouts per CDNA5 ISA (32-bit A 16x4): lanes 0-15 carry K=kk,kk+1,
// lanes 16-31 carry K=kk+2,kk+3 (one row per lane). B (4x16) mirrors it.
// D: VGPR j -> row (j + 8*half), col = lane&15.
// ---------------------------------------------------------------------------
__global__ void __launch_bounds__(128)
wmma_gemm_f32(const float* __restrict__ A, const float* __restrict__ Bm,
              float* __restrict__ D, int M, int tilesM)
{
    const int wave = blockIdx.x * 4 + threadIdx.y;
    const int totalTiles = tilesM * (H_DIM / 16);
    if (wave >= totalTiles) return;

    const int tm   = wave / (H_DIM / 16);
    const int tn   = wave % (H_DIM / 16);
    const int lane = threadIdx.x;
    const int half = lane >> 4;      // 0: K pair (kk,kk+1), 1: (kk+2,kk+3)
    const int l16  = lane & 15;
    const int row0 = tm * 16;
    const int col0 = tn * 16;

    int aRow = row0 + l16;
    if (aRow > M - 1) aRow = M - 1;  // clamp for M=16 / M=8 tables

    const float* __restrict__ aptr = A  + (size_t)aRow * K_DIM;
    const float* __restrict__ bcol = Bm + col0 + l16;

    v8f acc = {};
    for (int kk = 0; kk < K_DIM; kk += 4) {
        const int ka = kk + 2 * half;
        v2f afrag;
        afrag.x = aptr[ka];
        afrag.y = aptr[ka + 1];
        v2f bfrag;
        bfrag.x = bcol[(size_t)ka * H_DIM];
        bfrag.y = bcol[(size_t)(ka + 1) * H_DIM];
        acc = __builtin_amdgcn_wmma_f32_16x16x4_f32(false, afrag, false, bfrag,
                                                    (short)0, acc, false, false);
    }

#pragma unroll
    for (int j = 0; j < 8; ++j) {
        const int r = row0 + j + 8 * half;
        if (r < M) D[(size_t)r * H_DIM + col0 + l16] = acc[j];
    }
}

// ---------------------------------------------------------------------------
// Kernel 2: per-row syntax scan. One wave32 per batch row; each lane reduces
// a 64-token chunk to (delta-sum a, prefix-min m, seg-count c), a shuffle
// scan with the associative op (A1+A2, min(M1, A1+M2)) yields exclusive
// carries, then a second serial pass emits packed codes nest | (seg<<4).
// ---------------------------------------------------------------------------
__global__ void __launch_bounds__(32)
scan_codes(const int* __restrict__ tok, unsigned char* __restrict__ codes)
{
    const int b    = blockIdx.x;
    const int lane = threadIdx.x;
    const int CH   = S_DIM / 32;              // 64 tokens per lane
    const int base = b * S_DIM + lane * CH;
    const int BIG  = 0x3fffffff;

    int a = 0, m = BIG, c = 0;
    for (int i = 0; i < CH; ++i) {
        const int t = tok[base + i];
        const int d = ((t == 40) | (t == 123) | (t == 91))
                    - ((t == 41) | (t == 125) | (t == 93));
        a += d;
        m = min(m, a);
        c += (t > 39990) ? 1 : 0;
    }

    // Hillis-Steele inclusive scan of the non-commutative combine op.
    int ai = a, mi = m, ci = c;
    for (int off = 1; off < 32; off <<= 1) {
        const int au = __shfl_up(ai, off, 32);
        const int mu = __shfl_up(mi, off, 32);
        const int cu = __shfl_up(ci, off, 32);
        if (lane >= off) {
            mi = min(mu, au + mi);
            ai = au + ai;
            ci = cu + ci;
        }
    }
    int carryA = __shfl_up(ai, 1, 32);
    int carryM = __shfl_up(mi, 1, 32);
    int carryC = __shfl_up(ci, 1, 32);
    if (lane == 0) { carryA = 0; carryM = BIG; carryC = 0; }

    int s = carryA, runmin = carryM, segc = carryC;
    for (int i = 0; i < CH; ++i) {
        const int t = tok[base + i];
        const int d = ((t == 40) | (t == 123) | (t == 91))
                    - ((t == 41) | (t == 125) | (t == 93));
        s += d;
        runmin = min(runmin, s);
        int nest = s - min(runmin, 0);          // Lindley level
        nest = max(0, min(nest, 15));
        segc += (t > 39990) ? 1 : 0;
        const int seg = segc & 7;
        codes[base + i] = (unsigned char)(nest | (seg << 4));
    }
}

// ---------------------------------------------------------------------------
// Kernel 3: streaming emit, HBM-write bound (268 MB / 23.3 TB/s ~= 11.5 us).
// One block per (b,s); 128 threads x float4 cover H=512. P/N/G are
// L2-resident; prefetch hints keep upcoming P rows warm.
// ---------------------------------------------------------------------------
__global__ void __launch_bounds__(128)
emit_out(const float* __restrict__ P, const float* __restrict__ Nw,
         const float* __restrict__ Gw, const unsigned char* __restrict__ codes,
         float* __restrict__ out)
{
    const int bs = blockIdx.x;
    const int s  = bs & (S_DIM - 1);
    const int h  = threadIdx.x * 4;

    const unsigned code = codes[bs];
    const int nest = code & 15;
    const int seg  = (code >> 4) & 7;

    // gfx1250 global_prefetch_b8: hint the P row needed a few blocks ahead.
    __builtin_prefetch(P + (size_t)((s + 8) & (S_DIM - 1)) * H_DIM + h, 0, 1);

    const float4 p = *(const float4*)(P  + (size_t)s    * H_DIM + h);
    const float4 n = *(const float4*)(Nw + (size_t)nest * H_DIM + h);
    const float4 g = *(const float4*)(Gw + (size_t)seg  * H_DIM + h);

    float4 r;
    r.x = p.x + n.x + g.x;
    r.y = p.y + n.y + g.y;
    r.z = p.z + n.z + g.z;
    r.w = p.w + n.w + g.w;
    *(float4*)(out + (size_t)bs * H_DIM + h) = r;
}

// ---------------------------------------------------------------------------
extern "C" void kernel_launch(void* const* d_in, const int* in_sizes, int n_in,
                              void* d_out, int out_size, void* d_ws, size_t ws_size,
                              hipStream_t stream)
{
    const int*   tok  = (const int*)d_in[0];     // [B,S] int32
    const float* pos  = (const float*)d_in[1];   // [S,H]
    const float* nst  = (const float*)d_in[2];   // [16,H]
    const float* segt = (const float*)d_in[3];   // [8,H]
    const float* W    = (const float*)d_in[4];   // [3H,H]

    float* P  = (float*)d_ws;                    // [2048,512]  4 MB
    float* Nw = P  + (size_t)S_DIM * H_DIM;      // [16,512]
    float* Gw = Nw + (size_t)16 * H_DIM;         // [8,512]
    unsigned char* codes = (unsigned char*)(Gw + (size_t)8 * H_DIM); // [B,S]

    float* out = (float*)d_out;

    const dim3 blk(32, 4);  // 4 waves per block, one 16x16 tile each

    // P = pos_table @ W_pos
    {
        const int tilesM = S_DIM / 16;                 // 128
        const int tiles  = tilesM * (H_DIM / 16);      // 4096
        wmma_gemm_f32<<<(tiles + 3) / 4, blk, 0, stream>>>(pos, W, P, S_DIM, tilesM);
    }
    // N = nest_table @ W_nest   (W rows 512..1023)
    wmma_gemm_f32<<<8, blk, 0, stream>>>(nst, W + (size_t)512 * H_DIM, Nw, 16, 1);
    // G = seg_table @ W_seg     (W rows 1024..1535)
    wmma_gemm_f32<<<8, blk, 0, stream>>>(segt, W + (size_t)1024 * H_DIM, Gw, 8, 1);

    scan_codes<<<B_DIM, 32, 0, stream>>>(tok, codes);

    emit_out<<<B_DIM * S_DIM, 128, 0, stream>>>(P, Nw, Gw, codes, out);
}